// GAT_R_to_E_51634096833137
// MI455X (gfx1250) — compile-verified
//
#include <hip/hip_runtime.h>

#define E_HID 128
#define R_HID 128

typedef float v2f __attribute__((ext_vector_type(2)));
typedef float v8f __attribute__((ext_vector_type(8)));

__device__ __forceinline__ float leaky(float x) { return x > 0.f ? x : 0.01f * x; }

// float atomic max via sign-split integer atomics (init must be -inf = 0xFF800000)
__device__ __forceinline__ void atomicMaxF(float* addr, float val) {
  if (val >= 0.f) atomicMax((int*)addr, __float_as_int(val));
  else            atomicMin((unsigned int*)addr, __float_as_uint(val));
}

// ---------------- K0: init workspace ----------------
__global__ void k_init(float* __restrict__ x_acc, long n_acc,
                       float* __restrict__ mx_h, float* __restrict__ mx_t,
                       float* __restrict__ sm_h, float* __restrict__ sm_t, int n_e) {
  long i = (long)blockIdx.x * blockDim.x + threadIdx.x;
  long stride = (long)gridDim.x * blockDim.x;
  for (long j = i; j < n_acc; j += stride) x_acc[j] = 0.f;
  for (long j = i; j < n_e; j += stride) {
    mx_h[j] = -__builtin_inff(); mx_t[j] = -__builtin_inff();
    sm_h[j] = 0.f;               sm_t[j] = 0.f;
  }
}

// ---------------- K1: per-node attention scores (wave per row) ----------------
__global__ void k_node_scores(const float* __restrict__ x_e,
                              const float* __restrict__ w_ah,
                              const float* __restrict__ w_at,
                              float* __restrict__ e_h, float* __restrict__ e_t, int n_e) {
  int lane = threadIdx.x & 31;
  int wave = threadIdx.x >> 5;
  int row = blockIdx.x * 8 + wave;
  if (row >= n_e) return;
  const float4 x4 = ((const float4*)(x_e + (size_t)row * E_HID))[lane];
  const float4 a4 = ((const float4*)w_ah)[lane];
  const float4 b4 = ((const float4*)w_at)[lane];
  float sh = x4.x*a4.x + x4.y*a4.y + x4.z*a4.z + x4.w*a4.w;
  float st = x4.x*b4.x + x4.y*b4.y + x4.z*b4.z + x4.w*b4.w;
  for (int off = 16; off > 0; off >>= 1) {
    sh += __shfl_xor(sh, off);
    st += __shfl_xor(st, off);
  }
  if (lane == 0) { e_h[row] = sh; e_t[row] = st; }
}

// ---------------- K2: relation MLP + e_r (block per relation) ----------------
__global__ void k_rel(const float* __restrict__ x_r,
                      const float* __restrict__ w_ar,
                      const float* __restrict__ W1, const float* __restrict__ b1,
                      const float* __restrict__ W2, const float* __restrict__ b2,
                      float* __restrict__ e_r, float* __restrict__ msg, int n_r) {
  __shared__ float row[R_HID];
  __shared__ float hid[2 * R_HID];
  int r = blockIdx.x;
  int t = threadIdx.x;
  if (t < R_HID) row[t] = x_r[(size_t)r * R_HID + t];
  __syncthreads();
  // layer 1: hid[t] = b1[t] + sum_k row[k] * W1[k, t]   (W1 is [128, 256] row-major)
  float acc = b1[t];
  #pragma unroll 8
  for (int k = 0; k < R_HID; ++k) acc += row[k] * W1[k * 2 * R_HID + t];
  hid[t] = acc;
  // e_r: wave 0 reduces dot(row, w_ar)
  if (t < 32) {
    float s = 0.f;
    #pragma unroll
    for (int j = 0; j < 4; ++j) s += row[t + 32 * j] * w_ar[t + 32 * j];
    for (int off = 16; off > 0; off >>= 1) s += __shfl_xor(s, off);
    if (t == 0) e_r[r] = s;
  }
  __syncthreads();
  // layer 2 + residual: msg[t] = row[t] + b2[t] + sum_k hid[k] * W2[k, t]  (W2 is [256, 128])
  if (t < R_HID) {
    float o = b2[t];
    #pragma unroll 8
    for (int k = 0; k < 2 * R_HID; ++k) o += hid[k] * W2[k * R_HID + t];
    msg[(size_t)r * R_HID + t] = row[t] + o;
  }
}

// ---------------- K3: edge logits + segment max ----------------
__global__ void k_edge_max(const int* __restrict__ h_idx, const int* __restrict__ t_idx,
                           const int* __restrict__ rel,
                           const float* __restrict__ e_h, const float* __restrict__ e_t,
                           const float* __restrict__ e_r,
                           float* __restrict__ z_h, float* __restrict__ z_t,
                           float* __restrict__ mx_h, float* __restrict__ mx_t, int n_edges) {
  int e = blockIdx.x * blockDim.x + threadIdx.x;
  if (e >= n_edges) return;
  int h = h_idx[e], t = t_idx[e], r = rel[e];
  float er = e_r[r];
  float lh = leaky(e_h[h] + er);
  float lt = leaky(e_t[t] + er);
  z_h[e] = lh; z_t[e] = lt;
  atomicMaxF(mx_h + h, lh);
  atomicMaxF(mx_t + t, lt);
}

// ---------------- K4: exp + segment sum ----------------
__global__ void k_edge_sum(const int* __restrict__ h_idx, const int* __restrict__ t_idx,
                           const float* __restrict__ mx_h, const float* __restrict__ mx_t,
                           float* __restrict__ z_h, float* __restrict__ z_t,
                           float* __restrict__ sm_h, float* __restrict__ sm_t, int n_edges) {
  int e = blockIdx.x * blockDim.x + threadIdx.x;
  if (e >= n_edges) return;
  int h = h_idx[e], t = t_idx[e];
  float zh = __expf(z_h[e] - mx_h[h]);
  float zt = __expf(z_t[e] - mx_t[t]);
  z_h[e] = zh; z_t[e] = zt;
  atomicAdd(sm_h + h, zh);
  atomicAdd(sm_t + t, zt);
}

// ---------------- K5: weighted scatter-add (wave per edge) ----------------
__global__ void k_scatter(const int* __restrict__ h_idx, const int* __restrict__ t_idx,
                          const int* __restrict__ rel,
                          const float* __restrict__ z_h, const float* __restrict__ z_t,
                          const float* __restrict__ sm_h, const float* __restrict__ sm_t,
                          const float* __restrict__ msg, float* __restrict__ x_acc, int n_edges) {
  int gid = blockIdx.x * blockDim.x + threadIdx.x;
  int e = gid >> 5;
  int lane = gid & 31;
  if (e >= n_edges) return;
  int h = h_idx[e], t = t_idx[e], r = rel[e];
  float ah = z_h[e] / (sm_h[h] + 1e-16f);
  float at = z_t[e] / (sm_t[t] + 1e-16f);
  float4 m4 = ((const float4*)(msg + (size_t)r * R_HID))[lane];
  float* ph = x_acc + (size_t)h * 256 + lane * 4;
  atomicAdd(ph + 0, ah * m4.x); atomicAdd(ph + 1, ah * m4.y);
  atomicAdd(ph + 2, ah * m4.z); atomicAdd(ph + 3, ah * m4.w);
  float* pt = x_acc + (size_t)t * 256 + 128 + lane * 4;
  atomicAdd(pt + 0, at * m4.x); atomicAdd(pt + 1, at * m4.y);
  atomicAdd(pt + 2, at * m4.z); atomicAdd(pt + 3, at * m4.w);
}

// ---------------- K6: final GEMM [N_E,256] @ [256,256] + b via WMMA f32 ----------------
// Block: 256 threads (8 waves), covers 16 output rows x 256 cols.
// Each wave owns two 16x16 tiles: cols [w*16, w*16+16) and [w*16+128, ...).
// K staged in LDS chunks of 32.
//   A staged row-major  [16 rows][36]  -> per-lane fragment is one ds_load_b64
//   B staged TRANSPOSED [256 n][36 k]  -> per-lane fragment is one ds_load_b64 (no re-pack movs)
// Register double-buffering: next chunk's global loads issued before the WMMA loop.
__global__ void k_final_gemm(const float* __restrict__ x_acc,
                             const float* __restrict__ Wr, const float* __restrict__ br,
                             float* __restrict__ out, int n_e) {
  constexpr int KTOT = 256, NTOT = 256, KC = 32;
  constexpr int ASTR = 36, BKSTR = 36;           // padded strides (float4/float2-aligned)
  __shared__ float As[16 * ASTR];
  __shared__ float Bs[NTOT * BKSTR];             // Bs[n][k], transposed
  const int tid  = threadIdx.x;
  const int lane = tid & 31;
  const int wave = tid >> 5;
  const int m0   = blockIdx.x * 16;
  const int lrow = lane & 15;
  const int hsel = (lane >= 16) ? 2 : 0;         // A/B K-phase: lanes 16-31 hold K+2/K+3
  const int mrow = (lane >= 16) ? 8 : 0;         // C/D: lanes 16-31 hold M=8..15
  const int n0   = wave * 16;

  // A staging coords (threads 0..127): 16 rows x 32 k = 128 float4
  const int arow = tid >> 3;
  const int acol = (tid & 7) * 4;
  // B staging coords: thread tid owns column n = tid, loads k = i*4..i*4+3 for i in 0..7.
  // Each global_load is contiguous in n across lanes (coalesced 128B).

  v8f c0 = {}, c1 = {};
  float4 aReg = make_float4(0.f, 0.f, 0.f, 0.f);
  float4 bReg[8];

  auto loadGlobal = [&](int kb) {
    if (tid < 128) {
      int grow = m0 + arow;
      aReg = make_float4(0.f, 0.f, 0.f, 0.f);
      if (grow < n_e) aReg = *(const float4*)(x_acc + (size_t)grow * KTOT + kb + acol);
    }
    #pragma unroll
    for (int i = 0; i < 8; ++i) {
      int k4 = i * 4;
      bReg[i].x = Wr[(size_t)(kb + k4 + 0) * NTOT + tid];
      bReg[i].y = Wr[(size_t)(kb + k4 + 1) * NTOT + tid];
      bReg[i].z = Wr[(size_t)(kb + k4 + 2) * NTOT + tid];
      bReg[i].w = Wr[(size_t)(kb + k4 + 3) * NTOT + tid];
    }
  };
  auto storeSmem = [&]() {
    if (tid < 128) *(float4*)&As[arow * ASTR + acol] = aReg;
    #pragma unroll
    for (int i = 0; i < 8; ++i)
      *(float4*)&Bs[tid * BKSTR + i * 4] = bReg[i];
  };

  loadGlobal(0);
  for (int kb = 0; kb < KTOT; kb += KC) {
    storeSmem();
    __syncthreads();
    if (kb + KC < KTOT) loadGlobal(kb + KC);   // prefetch next chunk behind the WMMAs
    #pragma unroll
    for (int kk = 0; kk < KC; kk += 4) {
      int ka = kk + hsel;
      v2f a  = *(const v2f*)&As[lrow * ASTR + ka];
      v2f b0 = *(const v2f*)&Bs[(n0 + lrow) * BKSTR + ka];
      v2f b1 = *(const v2f*)&Bs[(n0 + 128 + lrow) * BKSTR + ka];
      c0 = __builtin_amdgcn_wmma_f32_16x16x4_f32(false, a, false, b0, (short)0, c0, false, false);
      c1 = __builtin_amdgcn_wmma_f32_16x16x4_f32(false, a, false, b1, (short)0, c1, false, false);
    }
    __syncthreads();
  }
  #pragma unroll
  for (int j = 0; j < 8; ++j) {
    int grow = m0 + mrow + j;
    if (grow < n_e) {
      out[(size_t)grow * NTOT + n0 + lrow]       = c0[j] + br[n0 + lrow];
      out[(size_t)grow * NTOT + n0 + 128 + lrow] = c1[j] + br[n0 + 128 + lrow];
    }
  }
}

extern "C" void kernel_launch(void* const* d_in, const int* in_sizes, int n_in,
                              void* d_out, int out_size, void* d_ws, size_t ws_size,
                              hipStream_t stream) {
  const float* x_e  = (const float*)d_in[0];
  const float* x_r  = (const float*)d_in[1];
  const int*   eidx = (const int*)d_in[2];
  const int*   rel  = (const int*)d_in[3];
  // d_in[4] = rel_all (unused by reference)
  const float* w_ah = (const float*)d_in[5];
  const float* w_at = (const float*)d_in[6];
  const float* w_ar = (const float*)d_in[7];
  const float* W1   = (const float*)d_in[8];
  const float* b1   = (const float*)d_in[9];
  const float* W2   = (const float*)d_in[10];
  const float* b2   = (const float*)d_in[11];
  const float* Wr   = (const float*)d_in[12];
  const float* br   = (const float*)d_in[13];
  float* out = (float*)d_out;

  const int n_e     = in_sizes[0] / E_HID;
  const int n_r     = in_sizes[1] / R_HID;
  const int n_edges = in_sizes[3];
  const int* h_idx = eidx;
  const int* t_idx = eidx + n_edges;

  // workspace carve (all 16B-aligned)
  float* x_acc = (float*)d_ws;                       // [n_e, 256]
  float* msg   = x_acc + (size_t)n_e * 256;          // [n_r, 128]
  float* z_h   = msg + (size_t)n_r * R_HID;          // [E]
  float* z_t   = z_h + n_edges;                      // [E]
  float* e_h   = z_t + n_edges;                      // [n_e]
  float* e_t   = e_h + n_e;                          // [n_e]
  float* e_r   = e_t + n_e;                          // [n_r]
  float* mx_h  = e_r + n_r;                          // [n_e]
  float* mx_t  = mx_h + n_e;                         // [n_e]
  float* sm_h  = mx_t + n_e;                         // [n_e]
  float* sm_t  = sm_h + n_e;                         // [n_e]

  k_init<<<4096, 256, 0, stream>>>(x_acc, (long)n_e * 256, mx_h, mx_t, sm_h, sm_t, n_e);
  k_node_scores<<<(n_e + 7) / 8, 256, 0, stream>>>(x_e, w_ah, w_at, e_h, e_t, n_e);
  k_rel<<<n_r, 256, 0, stream>>>(x_r, w_ar, W1, b1, W2, b2, e_r, msg, n_r);
  k_edge_max<<<(n_edges + 255) / 256, 256, 0, stream>>>(h_idx, t_idx, rel, e_h, e_t, e_r,
                                                        z_h, z_t, mx_h, mx_t, n_edges);
  k_edge_sum<<<(n_edges + 255) / 256, 256, 0, stream>>>(h_idx, t_idx, mx_h, mx_t,
                                                        z_h, z_t, sm_h, sm_t, n_edges);
  k_scatter<<<(n_edges + 7) / 8, 256, 0, stream>>>(h_idx, t_idx, rel, z_h, z_t,
                                                   sm_h, sm_t, msg, x_acc, n_edges);
  k_final_gemm<<<(n_e + 15) / 16, 256, 0, stream>>>(x_acc, Wr, br, out, n_e);
}